// MultiSimilarityLoss_16097537425965
// MI455X (gfx1250) — compile-verified
//
#include <hip/hip_runtime.h>
#include <math.h>

// MI455X / gfx1250, wave32. fp32 WMMA (V_WMMA_F32_16X16X4_F32).
typedef float v2f __attribute__((ext_vector_type(2)));
typedef float v8f __attribute__((ext_vector_type(8)));

#define B_SZ 8192
#define D_SZ 128
#define TILE 16
#define KSTEPS (D_SZ / 4)         // 32 wmma k-steps of K=4
#define WPB 4                     // waves per block
#define CSPLIT 4                  // column splits per row tile (parallelism)
#define CPW (B_SZ / CSPLIT)       // 2048 columns per wave
#define TPI 4                     // concurrent column tiles per iteration (ILP)

// 4 independent 16x16 sim tiles (shared A operand, 4 accumulator chains).
__device__ __forceinline__ void gemm4(const float* __restrict__ emb, const v2f* av,
                                      int cb, int hi, int lo, v8f acc[TPI]) {
    const float* bB[TPI];
#pragma unroll
    for (int t = 0; t < TPI; ++t) {
        bB[t] = emb + (size_t)(cb + t * TILE + lo) * D_SZ + 2 * hi;
        acc[t] = (v8f){0.f, 0.f, 0.f, 0.f, 0.f, 0.f, 0.f, 0.f};
    }
#pragma unroll
    for (int kk = 0; kk < KSTEPS; ++kk) {
        v2f q[TPI];
#pragma unroll
        for (int t = 0; t < TPI; ++t) q[t] = *(const v2f*)(bB[t] + 4 * kk);
#pragma unroll
        for (int t = 0; t < TPI; ++t)
            acc[t] = __builtin_amdgcn_wmma_f32_16x16x4_f32(
                false, av[kk], false, q[t], (short)0, acc[t], false, false);
    }
}

// Pass 1 (partial): per-row pos_min / neg_max over this wave's column range.
__global__ void __launch_bounds__(32 * WPB)
k_minmax_part(const float* __restrict__ emb, const int* __restrict__ labels,
              float* __restrict__ pmin_part, float* __restrict__ nmax_part) {
    const int wave = threadIdx.x >> 5;
    const int lane = threadIdx.x & 31;
    const int hi = lane >> 4, lo = lane & 15;
    const int gw = blockIdx.x * WPB + wave;     // 0 .. 2047
    const int rowBase = (gw / CSPLIT) * TILE;
    const int split = gw % CSPLIT;
    const int colBeg = split * CPW;

    const float* aBase = emb + (size_t)(rowBase + lo) * D_SZ + 2 * hi;
    v2f av[KSTEPS];
#pragma unroll
    for (int kk = 0; kk < KSTEPS; ++kk) av[kk] = *(const v2f*)(aBase + 4 * kk);

    int lrow[8];
#pragma unroll
    for (int i = 0; i < 8; ++i) lrow[i] = labels[rowBase + i + 8 * hi];

    float mn[8], mx[8];
#pragma unroll
    for (int i = 0; i < 8; ++i) { mn[i] = __builtin_inff(); mx[i] = -__builtin_inff(); }

    for (int cb = colBeg; cb < colBeg + CPW; cb += TPI * TILE) {
        v8f acc[TPI];
        gemm4(emb, av, cb, hi, lo, acc);
#pragma unroll
        for (int t = 0; t < TPI; ++t) {
            const int colBase = cb + t * TILE;
            const int lcol = labels[colBase + lo];
            const int gcol = colBase + lo;
#pragma unroll
            for (int i = 0; i < 8; ++i) {
                const int grow = rowBase + i + 8 * hi;
                const bool same = (lrow[i] == lcol);
                const float v = acc[t][i];
                mn[i] = (same && grow != gcol) ? fminf(mn[i], v) : mn[i];
                mx[i] = (!same) ? fmaxf(mx[i], v) : mx[i];
            }
        }
    }

    // butterfly across the 16 lanes holding each row (stays within half-wave)
#pragma unroll
    for (int m = 1; m <= 8; m <<= 1) {
#pragma unroll
        for (int i = 0; i < 8; ++i) {
            mn[i] = fminf(mn[i], __shfl_xor(mn[i], m, 32));
            mx[i] = fmaxf(mx[i], __shfl_xor(mx[i], m, 32));
        }
    }
    if (lo == 0) {
#pragma unroll
        for (int i = 0; i < 8; ++i) {
            const int r = rowBase + i + 8 * hi;
            pmin_part[split * B_SZ + r] = mn[i];
            nmax_part[split * B_SZ + r] = mx[i];
        }
    }
}

// Combine per-split min/max partials into full-row pos_min / neg_max.
__global__ void __launch_bounds__(256)
k_combine_minmax(const float* __restrict__ pmin_part, const float* __restrict__ nmax_part,
                 float* __restrict__ pos_min, float* __restrict__ neg_max) {
    const int r = blockIdx.x * 256 + threadIdx.x;
    if (r >= B_SZ) return;
    float mn = __builtin_inff(), mx = -__builtin_inff();
#pragma unroll
    for (int s = 0; s < CSPLIT; ++s) {
        mn = fminf(mn, pmin_part[s * B_SZ + r]);
        mx = fmaxf(mx, nmax_part[s * B_SZ + r]);
    }
    pos_min[r] = mn;
    neg_max[r] = mx;
}

// Pass 2 (partial): recompute sim tiles, hard mining, partial exp sums.
__global__ void __launch_bounds__(32 * WPB)
k_loss_part(const float* __restrict__ emb, const int* __restrict__ labels,
            const float* __restrict__ pos_min, const float* __restrict__ neg_max,
            float* __restrict__ ps_part, float* __restrict__ ns_part) {
    const int wave = threadIdx.x >> 5;
    const int lane = threadIdx.x & 31;
    const int hi = lane >> 4, lo = lane & 15;
    const int gw = blockIdx.x * WPB + wave;
    const int rowBase = (gw / CSPLIT) * TILE;
    const int split = gw % CSPLIT;
    const int colBeg = split * CPW;

    const float* aBase = emb + (size_t)(rowBase + lo) * D_SZ + 2 * hi;
    v2f av[KSTEPS];
#pragma unroll
    for (int kk = 0; kk < KSTEPS; ++kk) av[kk] = *(const v2f*)(aBase + 4 * kk);

    int lrow[8];
    float pm[8], nm[8], ps[8], ns[8];
#pragma unroll
    for (int i = 0; i < 8; ++i) {
        const int r = rowBase + i + 8 * hi;
        lrow[i] = labels[r];
        pm[i] = pos_min[r];
        nm[i] = neg_max[r];
        ps[i] = 0.f;
        ns[i] = 0.f;
    }

    for (int cb = colBeg; cb < colBeg + CPW; cb += TPI * TILE) {
        v8f acc[TPI];
        gemm4(emb, av, cb, hi, lo, acc);
#pragma unroll
        for (int t = 0; t < TPI; ++t) {
            const int colBase = cb + t * TILE;
            const int lcol = labels[colBase + lo];
            const int gcol = colBase + lo;
#pragma unroll
            for (int i = 0; i < 8; ++i) {
                const int grow = rowBase + i + 8 * hi;
                const bool same = (lrow[i] == lcol);
                const float v = acc[t][i];
                const bool hp = same && (grow != gcol) && (v - 0.1f < nm[i]);
                const bool hn = (!same) && (v + 0.1f > pm[i]);
                ps[i] += hp ? __expf(-2.0f * (v - 0.5f)) : 0.f;   // ALPHA=2, BASE=0.5
                ns[i] += hn ? __expf(50.0f * (v - 0.5f)) : 0.f;   // BETA=50
            }
        }
    }

#pragma unroll
    for (int m = 1; m <= 8; m <<= 1) {
#pragma unroll
        for (int i = 0; i < 8; ++i) {
            ps[i] += __shfl_xor(ps[i], m, 32);
            ns[i] += __shfl_xor(ns[i], m, 32);
        }
    }
    if (lo == 0) {
#pragma unroll
        for (int i = 0; i < 8; ++i) {
            const int r = rowBase + i + 8 * hi;
            ps_part[split * B_SZ + r] = ps[i];
            ns_part[split * B_SZ + r] = ns[i];
        }
    }
}

// Final: combine exp-sum partials, row loss + validity, scalar reduction.
__global__ void __launch_bounds__(256)
k_final(const float* __restrict__ ps_part, const float* __restrict__ ns_part,
        float* __restrict__ out) {
    __shared__ float sl[256];
    __shared__ float sv[256];
    float l = 0.f, v = 0.f;
    for (int r = threadIdx.x; r < B_SZ; r += 256) {
        float ps = 0.f, ns = 0.f;
#pragma unroll
        for (int s = 0; s < CSPLIT; ++s) {
            ps += ps_part[s * B_SZ + r];
            ns += ns_part[s * B_SZ + r];
        }
        // hard subsets of masks + exp>0  =>  valid  <=>  ps>0 && ns>0
        const bool valid = (ps > 0.f) && (ns > 0.f);
        l += valid ? (0.5f * log1pf(ps) + 0.02f * log1pf(ns)) : 0.f;
        v += valid ? 1.f : 0.f;
    }
    sl[threadIdx.x] = l;
    sv[threadIdx.x] = v;
    __syncthreads();
    for (int s = 128; s > 0; s >>= 1) {
        if (threadIdx.x < s) {
            sl[threadIdx.x] += sl[threadIdx.x + s];
            sv[threadIdx.x] += sv[threadIdx.x + s];
        }
        __syncthreads();
    }
    if (threadIdx.x == 0) out[0] = sl[0] / fmaxf(sv[0], 1.f);
}

extern "C" void kernel_launch(void* const* d_in, const int* in_sizes, int n_in,
                              void* d_out, int out_size, void* d_ws, size_t ws_size,
                              hipStream_t stream) {
    const float* emb = (const float*)d_in[0];   // (B, D) fp32, L2-normalized
    const int* labels = (const int*)d_in[1];    // (B,) int32

    float* pmin_part = (float*)d_ws;                     // [CSPLIT][B]
    float* nmax_part = pmin_part + CSPLIT * B_SZ;        // [CSPLIT][B]
    float* pos_min   = nmax_part + CSPLIT * B_SZ;        // [B]
    float* neg_max   = pos_min + B_SZ;                   // [B]
    float* ps_part   = neg_max + B_SZ;                   // [CSPLIT][B]
    float* ns_part   = ps_part + CSPLIT * B_SZ;          // [CSPLIT][B]
    // total: (4*CSPLIT + 2) * B_SZ * 4B = 576 KB workspace

    const int nWaves = (B_SZ / TILE) * CSPLIT;           // 2048 waves
    dim3 gridP(nWaves / WPB);                            // 512 blocks
    dim3 blkP(32 * WPB);                                 // 4 waves / block

    k_minmax_part<<<gridP, blkP, 0, stream>>>(emb, labels, pmin_part, nmax_part);
    k_combine_minmax<<<B_SZ / 256, 256, 0, stream>>>(pmin_part, nmax_part, pos_min, neg_max);
    k_loss_part<<<gridP, blkP, 0, stream>>>(emb, labels, pos_min, neg_max, ps_part, ns_part);
    k_final<<<1, 256, 0, stream>>>(ps_part, ns_part, (float*)d_out);
}